// SGC_24919400252015
// MI455X (gfx1250) — compile-verified
//
#include <hip/hip_runtime.h>

// ---------------------------------------------------------------------------
// SGC graph conv for gfx1250 (MI455X).
//   out = D_in^{-1/2} A^T D_out^{-1/2} (A^T)^2 (X @ W) + b
// (projection commuted to the front: hops run on 64-dim data, halving traffic)
// ---------------------------------------------------------------------------

typedef __attribute__((ext_vector_type(2))) float v2f;
typedef __attribute__((ext_vector_type(4))) float v4f;
typedef __attribute__((ext_vector_type(8))) float v8f;

#define IN_F  128
#define OUT_F 64

// ---------------- utility kernels ----------------

__global__ void zero_f32(float* __restrict__ p, int n) {
  int i = blockIdx.x * blockDim.x + threadIdx.x;
  int stride = gridDim.x * blockDim.x;
  for (; i < n; i += stride) p[i] = 0.0f;
}

__global__ void degree_kernel(const int* __restrict__ src,
                              const int* __restrict__ dst,
                              float* __restrict__ deg_out,   // out-degree of src nodes
                              float* __restrict__ deg_in,    // in-degree of dst nodes
                              int E) {
  int e = blockIdx.x * blockDim.x + threadIdx.x;
  if (e < E) {
    atomicAdd(&deg_out[src[e]], 1.0f);
    atomicAdd(&deg_in[dst[e]], 1.0f);
  }
}

// deg -> rsqrt(max(deg,1))  (applied to both norm arrays, laid out contiguously)
__global__ void norm_kernel(float* __restrict__ d, int n) {
  int i = blockIdx.x * blockDim.x + threadIdx.x;
  if (i < n) d[i] = rsqrtf(fmaxf(d[i], 1.0f));
}

// ---------------- fp32 WMMA GEMM: Y[N,64] = X[N,128] @ W[128,64] ----------------
// One wave per 16-row strip; 4 accumulators cover all 64 output columns.
// V_WMMA_F32_16X16X4_F32: A 16x4 (2 VGPR/lane), B 4x16 (2 VGPR/lane), C/D 16x16 (8 VGPR).

__global__ __launch_bounds__(128)
void gemm_xw_wmma(const float* __restrict__ X, const float* __restrict__ W,
                  float* __restrict__ Y, int n_tiles, int N) {
  const int wave = blockIdx.x * (blockDim.x >> 5) + (threadIdx.x >> 5);
  if (wave >= n_tiles) return;            // wave-uniform: EXEC stays all-1s for WMMA

  const int lane = threadIdx.x & 31;
  const int half = lane >> 4;             // 0: K pair {0,1}, 1: K pair {2,3}
  const int l16  = lane & 15;

  // A layout (16x4 f32): lane<16 -> M=lane, VGPR0=K0, VGPR1=K1; lane>=16 -> K2,K3
  int arow = wave * 16 + l16;
  if (arow >= N) arow = N - 1;            // clamp loads; stores guarded below
  const float* xrow = X + (long)arow * IN_F;

  v8f acc0 = {}, acc1 = {}, acc2 = {}, acc3 = {};

  for (int kk = 0; kk < IN_F; kk += 4) {
    const int ka = kk + half * 2;
    v2f A;
    A.x = xrow[ka];
    A.y = xrow[ka + 1];

    // B layout (4x16 f32): lane<16 -> N=lane, VGPR0=K0, VGPR1=K1; lane>=16 -> K2,K3
    const float* wrow = W + ka * OUT_F + l16;
    v2f B0, B1, B2, B3;
    B0.x = wrow[0];   B0.y = wrow[OUT_F + 0];
    B1.x = wrow[16];  B1.y = wrow[OUT_F + 16];
    B2.x = wrow[32];  B2.y = wrow[OUT_F + 32];
    B3.x = wrow[48];  B3.y = wrow[OUT_F + 48];

    acc0 = __builtin_amdgcn_wmma_f32_16x16x4_f32(false, A, false, B0, (short)0, acc0, false, false);
    acc1 = __builtin_amdgcn_wmma_f32_16x16x4_f32(false, A, false, B1, (short)0, acc1, false, false);
    acc2 = __builtin_amdgcn_wmma_f32_16x16x4_f32(false, A, false, B2, (short)0, acc2, false, false);
    acc3 = __builtin_amdgcn_wmma_f32_16x16x4_f32(false, A, false, B3, (short)0, acc3, false, false);
  }

  // C/D layout: VGPR v, lanes 0-15 -> M=v, lanes 16-31 -> M=v+8; N=l16
  const int rbase = wave * 16 + half * 8;
#pragma unroll
  for (int v = 0; v < 8; ++v) {
    const int r = rbase + v;
    if (r < N) {
      float* p = Y + (long)r * OUT_F + l16;
      p[0]  = acc0[v];
      p[16] = acc1[v];
      p[32] = acc2[v];
      p[48] = acc3[v];
    }
  }
}

// ---------------- edge-parallel hop: out[dst] += in[src] * (scale ? scale[src] : 1) ----
// One thread per (edge, 4-float chunk): 16 consecutive threads cover one 64-dim row,
// so the float4 gather and the 4 global_atomic_add_f32 are fully coalesced.

__global__ __launch_bounds__(256)
void hop_kernel(const float* __restrict__ in, float* __restrict__ out,
                const int* __restrict__ src, const int* __restrict__ dst,
                const float* __restrict__ scale, int E) {
  const long t = (long)blockIdx.x * blockDim.x + threadIdx.x;
  const long total = (long)E * (OUT_F / 4);
  if (t >= total) return;
  const int e = (int)(t >> 4);            // t / 16
  const int q = ((int)t & 15) << 2;       // float4 chunk within the row
  const int s = src[e];
  const int d = dst[e];

  v4f val = *(const v4f*)(in + (long)s * OUT_F + q);
  const float sc = scale ? scale[s] : 1.0f;

  float* o = out + (long)d * OUT_F + q;
  atomicAdd(o + 0, val.x * sc);
  atomicAdd(o + 1, val.y * sc);
  atomicAdd(o + 2, val.z * sc);
  atomicAdd(o + 3, val.w * sc);
}

// ---------------- finalize: out[i] = out[i]*norm_dst[row] + bias[col] (in place) -------

__global__ void finalize_kernel(float* __restrict__ out,
                                const float* __restrict__ norm_dst,
                                const float* __restrict__ bias, int n) {
  int i = blockIdx.x * blockDim.x + threadIdx.x;
  if (i < n) out[i] = out[i] * norm_dst[i >> 6] + bias[i & 63];
}

// ---------------------------------------------------------------------------

extern "C" void kernel_launch(void* const* d_in, const int* in_sizes, int n_in,
                              void* d_out, int out_size, void* d_ws, size_t ws_size,
                              hipStream_t stream) {
  const float* X    = (const float*)d_in[0];   // [N,128]
  const int*   src  = (const int*)d_in[1];     // [E]
  const int*   dst  = (const int*)d_in[2];     // [E]
  const float* W    = (const float*)d_in[3];   // [128,64]
  const float* bias = (const float*)d_in[4];   // [64]
  float* out = (float*)d_out;                  // [N,64]

  const int N = in_sizes[0] / IN_F;
  const int E = in_sizes[1];

  // workspace layout (floats): [norm_src N][norm_dst N][Y0 N*64][Y1 N*64][Y2 N*64]
  float* ws       = (float*)d_ws;
  float* norm_src = ws;
  float* norm_dst = ws + N;
  float* Y0 = ws + 2 * (size_t)N;
  float* Y1 = Y0 + (size_t)N * OUT_F;
  float* Y2 = Y1 + (size_t)N * OUT_F;

  const int yn = N * OUT_F;
  const int zb = 256;

  // zero scatter destinations (degrees, Y1, Y2, out)
  zero_f32<<<(2 * N + zb - 1) / zb, zb, 0, stream>>>(norm_src, 2 * N);
  zero_f32<<<(yn + zb - 1) / zb, zb, 0, stream>>>(Y1, yn);
  zero_f32<<<(yn + zb - 1) / zb, zb, 0, stream>>>(Y2, yn);
  zero_f32<<<(yn + zb - 1) / zb, zb, 0, stream>>>(out, yn);

  // degrees -> norms (both arrays contiguous)
  degree_kernel<<<(E + zb - 1) / zb, zb, 0, stream>>>(src, dst, norm_src, norm_dst, E);
  norm_kernel<<<(2 * N + zb - 1) / zb, zb, 0, stream>>>(norm_src, 2 * N);

  // Y0 = X @ W  (fp32 WMMA; projection moved ahead of the hops)
  const int tiles = (N + 15) / 16;
  gemm_xw_wmma<<<(tiles + 3) / 4, 128, 0, stream>>>(X, W, Y0, tiles, N);

  // three aggregation hops on 64-dim data (norm_src folded into the last gather)
  const long chunks = (long)E * (OUT_F / 4);
  const int hblocks = (int)((chunks + 255) / 256);
  hop_kernel<<<hblocks, 256, 0, stream>>>(Y0, Y1, src, dst, nullptr, E);
  hop_kernel<<<hblocks, 256, 0, stream>>>(Y1, Y2, src, dst, nullptr, E);
  hop_kernel<<<hblocks, 256, 0, stream>>>(Y2, out, src, dst, norm_src, E);

  // out = out * norm_dst + bias
  finalize_kernel<<<(yn + zb - 1) / zb, zb, 0, stream>>>(out, norm_dst, bias, yn);
}